// KeypointOnPCLoss_30992484008034
// MI455X (gfx1250) — compile-verified
//
#include <hip/hip_runtime.h>
#include <hip/hip_bf16.h>

typedef float v2f __attribute__((ext_vector_type(2)));
typedef float v8f __attribute__((ext_vector_type(8)));

#define BATCH 16
#define MDIM  512
#define NDIM  16384
#define WAVES 8          // 8 wave32s = 256 threads per block
#define NTILES (NDIM / 16)

// One block per (batch, m-tile of 16 keypoints). Each wave scans an
// interleaved subset of the 1024 n-tiles with V_WMMA_F32_16X16X4_F32,
// tracking per-row (min score, argmin index) in registers.
// Score[m][n] = -2 k.p + |p|^2 (same argmin as squared distance).
__global__ __launch_bounds__(256)
void keypoint_on_pc_loss_kernel(const float* __restrict__ kp,   // (B,3,M)
                                const float* __restrict__ pc,   // (B,3,N)
                                const float* __restrict__ sn,   // (B,3,N)
                                float* __restrict__ out)        // (B,M)
{
    const int blk  = blockIdx.x;          // 0..511
    const int b    = blk >> 5;            // / (M/16)
    const int mt   = blk & 31;
    const int tid  = threadIdx.x;
    const int wave = tid >> 5;
    const int lane = tid & 31;
    const int ln   = lane & 15;
    const bool hi  = lane >= 16;

    const float* kb = kp + (size_t)b * 3 * MDIM;
    const float* pb = pc + (size_t)b * 3 * NDIM;
    const int m0 = mt * 16;

    // ---- Build A (16x4 fp32): row m = (-2kx, -2ky, -2kz, 1) ----
    // Lane layout: lanes 0-15 hold {K0,K1} for M=ln; lanes 16-31 hold {K2,K3}.
    const float kx = kb[0 * MDIM + m0 + ln];
    const float ky = kb[1 * MDIM + m0 + ln];
    const float kz = kb[2 * MDIM + m0 + ln];
    v2f amat;
    amat.x = hi ? (-2.0f * kz) : (-2.0f * kx);
    amat.y = hi ? 1.0f         : (-2.0f * ky);

    // Running per-lane argmin for 8 D rows (rows 0-7 for lanes<16, 8-15 else).
    float minv[8];
    int   mini[8];
#pragma unroll
    for (int r = 0; r < 8; ++r) { minv[r] = 1.0e30f; mini[r] = 0; }

    // ---- Scan n-tiles, 2 tiles per iteration (independent WMMAs in flight).
    // NTILES/WAVES = 128 tiles per wave -> 64 double-iterations, no remainder.
    for (int t = wave; t < NTILES; t += 2 * WAVES) {
        const int n0 = t * 16 + ln;
        const int n1 = n0 + WAVES * 16;

        const float px0 = pb[0 * NDIM + n0];
        const float py0 = pb[1 * NDIM + n0];
        const float pz0 = pb[2 * NDIM + n0];
        const float px1 = pb[0 * NDIM + n1];
        const float py1 = pb[1 * NDIM + n1];
        const float pz1 = pb[2 * NDIM + n1];

        // B (4x16): lanes 0-15 hold {K0,K1}={px,py} for N=ln;
        //           lanes 16-31 hold {K2,K3}={pz,|p|^2}.
        v2f b0, b1;
        b0.x = hi ? pz0 : px0;
        b0.y = hi ? (px0 * px0 + py0 * py0 + pz0 * pz0) : py0;
        b1.x = hi ? pz1 : px1;
        b1.y = hi ? (px1 * px1 + py1 * py1 + pz1 * pz1) : py1;

        v8f acc0 = {};
        v8f acc1 = {};
        acc0 = __builtin_amdgcn_wmma_f32_16x16x4_f32(
            false, amat, false, b0, (short)0, acc0, false, false);
        acc1 = __builtin_amdgcn_wmma_f32_16x16x4_f32(
            false, amat, false, b1, (short)0, acc1, false, false);

#pragma unroll
        for (int r = 0; r < 8; ++r) {
            const float d0 = acc0[r];
            if (d0 < minv[r]) { minv[r] = d0; mini[r] = n0; }
        }
#pragma unroll
        for (int r = 0; r < 8; ++r) {
            const float d1 = acc1[r];
            if (d1 < minv[r]) { minv[r] = d1; mini[r] = n1; }
        }
    }

    // ---- Butterfly reduce across the 16 lanes of each half (branchless) ----
#pragma unroll
    for (int mask = 1; mask < 16; mask <<= 1) {
#pragma unroll
        for (int r = 0; r < 8; ++r) {
            const float ov = __shfl_xor(minv[r], mask, 32);
            const int   oi = __shfl_xor(mini[r], mask, 32);
            const bool take = (ov < minv[r]) | ((ov == minv[r]) & (oi < mini[r]));
            minv[r] = take ? ov : minv[r];
            mini[r] = take ? oi : mini[r];
        }
    }

    // ---- Cross-wave reduction in LDS ----
    __shared__ float s_val[WAVES][16];
    __shared__ int   s_idx[WAVES][16];
    if (lane == 0) {
#pragma unroll
        for (int r = 0; r < 8; ++r) { s_val[wave][r] = minv[r]; s_idx[wave][r] = mini[r]; }
    } else if (lane == 16) {
#pragma unroll
        for (int r = 0; r < 8; ++r) { s_val[wave][8 + r] = minv[r]; s_idx[wave][8 + r] = mini[r]; }
    }
    __syncthreads();

    // ---- Epilogue: 16 lanes each finish one keypoint row ----
    if (tid < 16) {
        float bv = s_val[0][tid];
        int   bi = s_idx[0][tid];
#pragma unroll
        for (int w = 1; w < WAVES; ++w) {
            const float v = s_val[w][tid];
            const int   i = s_idx[w][tid];
            const bool take = (v < bv) | ((v == bv) & (i < bi));
            bv = take ? v : bv;
            bi = take ? i : bi;
        }
        const int m = m0 + tid;
        const float qx = kb[0 * MDIM + m];
        const float qy = kb[1 * MDIM + m];
        const float qz = kb[2 * MDIM + m];
        const float px = pb[0 * NDIM + bi];
        const float py = pb[1 * NDIM + bi];
        const float pz = pb[2 * NDIM + bi];
        const float* sb = sn + (size_t)b * 3 * NDIM;
        const float sx = sb[0 * NDIM + bi];
        const float sy = sb[1 * NDIM + bi];
        const float sz = sb[2 * NDIM + bi];

        const float dx = qx - px, dy = qy - py, dz = qz - pz;
        const float nrm = sqrtf(dx * dx + dy * dy + dz * dz);
        const float inv = 1.0f / (nrm + 1.0e-7f);
        const float dot = (sx * dx + sy * dy + sz * dz) * inv;
        out[(size_t)b * MDIM + m] = dot * dot;
    }
}

extern "C" void kernel_launch(void* const* d_in, const int* in_sizes, int n_in,
                              void* d_out, int out_size, void* d_ws, size_t ws_size,
                              hipStream_t stream) {
    (void)in_sizes; (void)n_in; (void)d_ws; (void)ws_size; (void)out_size;
    const float* kp = (const float*)d_in[0];  // (16,3,512)
    const float* pc = (const float*)d_in[1];  // (16,3,16384)
    const float* sn = (const float*)d_in[2];  // (16,3,16384)
    float* out = (float*)d_out;               // (16,512,1,1) flat

    dim3 grid(BATCH * (MDIM / 16));           // 512 blocks
    dim3 block(32 * WAVES);                   // 256 threads = 8 wave32
    keypoint_on_pc_loss_kernel<<<grid, block, 0, stream>>>(kp, pc, sn, out);
}